// linearBasisConvLT_52261162057778
// MI455X (gfx1250) — compile-verified
//
#include <hip/hip_runtime.h>

typedef __attribute__((ext_vector_type(2))) float v2f;
typedef __attribute__((ext_vector_type(8))) float v8f;

#define B_BATCH 2
#define BB 16
#define IC 8
#define OC 16
#define QQ 30
#define MMm 6
#define UU 5
#define VV 5
#define KC (IC * UU * VV)   // 200 complex K
#define TILES_PER_QM 32     // (BATCH*BB*BB)/16 = 512/16

// One wave (32 threads) per block. Block = one 16x16 output tile of one (q,m) slice.
__global__ __launch_bounds__(32)
void linearBasisConvLT_kernel(
    const float* __restrict__ x_re, const float* __restrict__ x_im,
    const float* __restrict__ lnAlpha, const float* __restrict__ Phi,
    const float* __restrict__ lnTau,  const float* __restrict__ Psi,
    const float* __restrict__ pAlpha_re, const float* __restrict__ pAlpha_im,
    const float* __restrict__ pPhi_re,   const float* __restrict__ pPhi_im,
    const float* __restrict__ pTau_re,   const float* __restrict__ pTau_im,
    const float* __restrict__ pPsi_re,   const float* __restrict__ pPsi_im,
    const float* __restrict__ F_re,      const float* __restrict__ F_im,
    const float* __restrict__ coeffs,
    float* __restrict__ out)
{
    __shared__ float Gre[16][KC + 1];   // +1 pad: stride 201 -> conflict-free A-frag reads
    __shared__ float Gim[16][KC + 1];
    __shared__ float Hre[KC][16];
    __shared__ float Him[KC][16];

    const int lane = threadIdx.x;            // 0..31 (wave32)
    const int tile = blockIdx.x & (TILES_PER_QM - 1);
    const int qm   = blockIdx.x / TILES_PER_QM;   // 0..179
    const int q = qm / MMm, m = qm % MMm;
    const int b = tile >> 4, y = tile & 15;

    // ---- Build H[(c,u,v), o] = F[q,m,u,v] * Wr[c,u,v,o] for this (q,m) slice ----
    // Wr from collect_weights(coeffs): j = u*5+v in 0..24
    //   j==12 -> (coeffs[c,0,o], 0); j<12 -> t=11-j; j>12 -> t=j-13;
    //   w = coeffs[c,1+2t,o] + i*coeffs[c,2+2t,o]
    for (int i = lane; i < KC * 16; i += 32) {
        const int o = i & 15;
        const int k = i >> 4;        // c*25 + j
        const int c = k / 25;
        const int j = k % 25;
        float wre, wim;
        if (j == 12) {
            wre = coeffs[(c * 25 + 0) * OC + o];
            wim = 0.0f;
        } else {
            const int t = (j < 12) ? (11 - j) : (j - 13);
            wre = coeffs[(c * 25 + 1 + 2 * t) * OC + o];
            wim = coeffs[(c * 25 + 2 + 2 * t) * OC + o];
        }
        const float fre = F_re[qm * 25 + j];
        const float fim = F_im[qm * 25 + j];
        Hre[k][o] = fre * wre - fim * wim;
        Him[k][o] = fre * wim + fim * wre;
    }

    // ---- Per-(q,m) complex parameters ----
    const float paRe = pAlpha_re[qm], paIm = pAlpha_im[qm];
    const float ppRe = pPhi_re[m],    ppIm = pPhi_im[m];
    float ptRe[UU], ptIm[UU], psRe[VV], psIm[VV];
    #pragma unroll
    for (int u = 0; u < UU; ++u) { ptRe[u] = pTau_re[qm * UU + u]; ptIm[u] = pTau_im[qm * UU + u]; }
    #pragma unroll
    for (int v = 0; v < VV; ++v) { psRe[v] = pPsi_re[m * VV + v]; psIm[v] = pPsi_im[m * VV + v]; }

    // ---- Build G tile: 16 rows (x) x 8 channels -> 128 (row,c) pairs, 4 per lane ----
    #pragma unroll
    for (int i = 0; i < 4; ++i) {
        const int p  = lane + 32 * i;     // 0..127
        const int xr = p >> 3;            // row within tile = x coordinate
        const int c  = p & 7;
        const int gidx = (((b * BB + y) * BB + xr) * IC) + c;
        const float la = lnAlpha[gidx], ph = Phi[gidx];
        const float lt = lnTau[gidx],   ps = Psi[gidx];
        const float xre = x_re[gidx],   xim = x_im[gidx];

        // A = x * exp(la*pAlpha + ph*pPhi)
        const float er = la * paRe + ph * ppRe;
        const float ei = la * paIm + ph * ppIm;
        const float mag = __expf(er);
        float sn, cs; __sincosf(ei, &sn, &cs);
        const float Are = mag * (xre * cs - xim * sn);
        const float Aim = mag * (xre * sn + xim * cs);

        float EvRe[VV], EvIm[VV];
        #pragma unroll
        for (int v = 0; v < VV; ++v) {
            const float m2 = __expf(ps * psRe[v]);
            float s2, c2; __sincosf(ps * psIm[v], &s2, &c2);
            EvRe[v] = m2 * c2; EvIm[v] = m2 * s2;
        }
        #pragma unroll
        for (int u = 0; u < UU; ++u) {
            const float m1 = __expf(lt * ptRe[u]);
            float s1, c1; __sincosf(lt * ptIm[u], &s1, &c1);
            const float eur = m1 * c1, eui = m1 * s1;
            const float AuRe = Are * eur - Aim * eui;
            const float AuIm = Are * eui + Aim * eur;
            #pragma unroll
            for (int v = 0; v < VV; ++v) {
                const int k = c * 25 + u * 5 + v;
                Gre[xr][k] = AuRe * EvRe[v] - AuIm * EvIm[v];
                Gim[xr][k] = AuRe * EvIm[v] + AuIm * EvRe[v];
            }
        }
    }
    __syncthreads();

    // ---- Complex GEMM via V_WMMA_F32_16X16X4_F32 (f32 in, f32 acc) ----
    // A-frag: lane%16 = row M, lane/16 selects K pair {2h, 2h+1} in the 2 VGPRs.
    // B-frag mirrored: lane%16 = col N, lane/16 selects K pair.
    v8f accRe = {};
    v8f accIm = {};
    const int rowA  = lane & 15;
    const int khalf = (lane >> 4) << 1;       // 0 or 2
    for (int k0 = 0; k0 < KC; k0 += 4) {
        const int ka = k0 + khalf;
        v2f aRe, aIm, bRe, bIm, nbIm;
        aRe.x = Gre[rowA][ka];     aRe.y = Gre[rowA][ka + 1];
        aIm.x = Gim[rowA][ka];     aIm.y = Gim[rowA][ka + 1];
        bRe.x = Hre[ka][rowA];     bRe.y = Hre[ka + 1][rowA];
        bIm.x = Him[ka][rowA];     bIm.y = Him[ka + 1][rowA];
        nbIm.x = -bIm.x;           nbIm.y = -bIm.y;
        // out_re += Gre*Hre - Gim*Him ; out_im += Gre*Him + Gim*Hre
        accRe = __builtin_amdgcn_wmma_f32_16x16x4_f32(false, aRe, false, bRe,  (short)0, accRe, false, false);
        accRe = __builtin_amdgcn_wmma_f32_16x16x4_f32(false, aIm, false, nbIm, (short)0, accRe, false, false);
        accIm = __builtin_amdgcn_wmma_f32_16x16x4_f32(false, aRe, false, bIm,  (short)0, accIm, false, false);
        accIm = __builtin_amdgcn_wmma_f32_16x16x4_f32(false, aIm, false, bRe,  (short)0, accIm, false, false);
    }

    // ---- Store: C/D layout -> VGPR vi, lane: row M = vi + 8*(lane/16), col N = lane%16.
    // out[b, o, q, m, y, x, {re,im}]
    const int o = lane & 15;
    const int xbase = (lane >> 4) * 8;
    #pragma unroll
    for (int vi = 0; vi < 8; ++vi) {
        const int xo = xbase + vi;
        const long flat =
            (((((long)(b * OC + o) * QQ + q) * MMm + m) * BB + y) * BB + xo) * 2;
        float2 val;
        val.x = accRe[vi];
        val.y = accIm[vi];
        *(float2*)(out + flat) = val;  // flat is even -> 8B aligned
    }
}

extern "C" void kernel_launch(void* const* d_in, const int* in_sizes, int n_in,
                              void* d_out, int out_size, void* d_ws, size_t ws_size,
                              hipStream_t stream) {
    (void)in_sizes; (void)n_in; (void)d_ws; (void)ws_size; (void)out_size;
    const float* x_re      = (const float*)d_in[0];
    const float* x_im      = (const float*)d_in[1];
    const float* lnAlpha   = (const float*)d_in[2];
    const float* Phi       = (const float*)d_in[3];
    const float* lnTau     = (const float*)d_in[4];
    const float* Psi       = (const float*)d_in[5];
    const float* pAlpha_re = (const float*)d_in[6];
    const float* pAlpha_im = (const float*)d_in[7];
    const float* pPhi_re   = (const float*)d_in[8];
    const float* pPhi_im   = (const float*)d_in[9];
    const float* pTau_re   = (const float*)d_in[10];
    const float* pTau_im   = (const float*)d_in[11];
    const float* pPsi_re   = (const float*)d_in[12];
    const float* pPsi_im   = (const float*)d_in[13];
    const float* F_re      = (const float*)d_in[14];
    const float* F_im      = (const float*)d_in[15];
    const float* coeffs    = (const float*)d_in[16];
    float* out = (float*)d_out;

    const int nBlocks = QQ * MMm * TILES_PER_QM;  // 180 * 32 = 5760
    linearBasisConvLT_kernel<<<nBlocks, 32, 0, stream>>>(
        x_re, x_im, lnAlpha, Phi, lnTau, Psi,
        pAlpha_re, pAlpha_im, pPhi_re, pPhi_im,
        pTau_re, pTau_im, pPsi_re, pPsi_im,
        F_re, F_im, coeffs, out);
}